// BothSideAttention_13993003451113
// MI455X (gfx1250) — compile-verified
//
#include <hip/hip_runtime.h>
#include <math.h>

// Problem constants (reference: B=32, S=8192, H=512)
#define B_   32
#define S_   8192
#define H_   512
#define CPB  32                          // chunks per batch (split-K over S)
#define ROWS_PER_CHUNK (S_ / CPB)        // 256
#define WPG  8                           // waves per workgroup (256 threads, wave32)
#define ROWS_PER_WAVE (ROWS_PER_CHUNK / WPG) // 32

typedef __attribute__((ext_vector_type(2))) float v2f;
typedef __attribute__((ext_vector_type(8))) float v8f;

// Workspace layout (in floats):
//   [0, B*CPB*H)         : partial weighted accumulators acc[b][chunk][h]
//   [WS_M, +B*CPB)       : partial running max m
//   [WS_L, +B*CPB)       : partial denominators l
//   [WS_COMB, +B*4H)     : combined matrix [B][4H]
#define WS_ACC  0
#define WS_M    (B_ * CPB * H_)
#define WS_L    (WS_M + B_ * CPB)
#define WS_COMB (WS_L + B_ * CPB)

__device__ __forceinline__ float dot4(const float4& q, const float4& v) {
    return q.x * v.x + q.y * v.y + q.z * v.z + q.w * v.w;
}
__device__ __forceinline__ void fma4(float4& acc, float s, const float4& v) {
    acc.x += s * v.x; acc.y += s * v.y; acc.z += s * v.z; acc.w += s * v.w;
}
__device__ __forceinline__ void scl4(float4& acc, float s) {
    acc.x *= s; acc.y *= s; acc.z *= s; acc.w *= s;
}

// ---------------------------------------------------------------------------
// Kernel 1: single streaming pass over context.
//   - attn[b,s] = dot(output[b], context[b,s])   (written for ALL s)
//   - online-softmax accumulation of sum_s softmax(attn)*context over s<split
//   One wave processes whole rows (2KB each) with float4 loads; partials per
//   (batch, chunk) are merged across the 8 waves in LDS and spilled to ws.
// ---------------------------------------------------------------------------
__global__ __launch_bounds__(256)
void k1_stream(const float* __restrict__ outp, const float* __restrict__ ctx,
               const int* __restrict__ split, float* __restrict__ attn,
               float* __restrict__ ws) {
    __shared__ float sacc[WPG * H_];
    __shared__ float smx[WPG];
    __shared__ float sln[WPG];

    const int b     = blockIdx.y;
    const int chunk = blockIdx.x;
    const int tid   = threadIdx.x;
    const int wv    = tid >> 5;
    const int lane  = tid & 31;
    const int sbase = chunk * ROWS_PER_CHUNK;
    const int sp    = split[b];

    // Preload query row (512 floats -> 16 per lane as 4 x float4)
    const float* qp = outp + (size_t)b * H_;
    float4 q0 = ((const float4*)qp)[0 * 32 + lane];
    float4 q1 = ((const float4*)qp)[1 * 32 + lane];
    float4 q2 = ((const float4*)qp)[2 * 32 + lane];
    float4 q3 = ((const float4*)qp)[3 * 32 + lane];

    const float* cbase = ctx + (size_t)b * S_ * H_;

    float m = -3.0e38f, l = 0.0f;
    float4 a0 = {0,0,0,0}, a1 = {0,0,0,0}, a2 = {0,0,0,0}, a3 = {0,0,0,0};

    // Prime the register pipeline with this wave's first row.
    const int s0 = sbase + wv;
    const float* rp = cbase + (size_t)s0 * H_;
    float4 x0 = ((const float4*)rp)[0 * 32 + lane];
    float4 x1 = ((const float4*)rp)[1 * 32 + lane];
    float4 x2 = ((const float4*)rp)[2 * 32 + lane];
    float4 x3 = ((const float4*)rp)[3 * 32 + lane];

    for (int i = 0; i < ROWS_PER_WAVE; ++i) {
        const int s = sbase + wv + i * WPG;

        // Software-prefetch next row into registers (overlaps the reduction).
        float4 y0 = x0, y1 = x1, y2 = x2, y3 = x3;
        if (i + 1 < ROWS_PER_WAVE) {
            const float* np = cbase + (size_t)(s + WPG) * H_;
            y0 = ((const float4*)np)[0 * 32 + lane];
            y1 = ((const float4*)np)[1 * 32 + lane];
            y2 = ((const float4*)np)[2 * 32 + lane];
            y3 = ((const float4*)np)[3 * 32 + lane];
        }

        // Per-lane dot partial, then wave32 butterfly reduction.
        float p = dot4(q0, x0) + dot4(q1, x1) + dot4(q2, x2) + dot4(q3, x3);
        #pragma unroll
        for (int off = 16; off >= 1; off >>= 1)
            p += __shfl_xor(p, off, 32);

        if (lane == 0) attn[(size_t)b * S_ + s] = p;   // raw (unmasked) logits

        if (s < sp) {                                   // online softmax update
            const float mOld = m;
            m = fmaxf(m, p);
            if (m != mOld) {                            // rare rescale
                const float sc = __expf(mOld - m);
                l *= sc;
                scl4(a0, sc); scl4(a1, sc); scl4(a2, sc); scl4(a3, sc);
            }
            const float w = __expf(p - m);
            l += w;
            fma4(a0, w, x0); fma4(a1, w, x1); fma4(a2, w, x2); fma4(a3, w, x3);
        }

        x0 = y0; x1 = y1; x2 = y2; x3 = y3;
    }

    // Merge the 8 waves' partials through LDS.
    ((float4*)&sacc[wv * H_])[0 * 32 + lane] = a0;
    ((float4*)&sacc[wv * H_])[1 * 32 + lane] = a1;
    ((float4*)&sacc[wv * H_])[2 * 32 + lane] = a2;
    ((float4*)&sacc[wv * H_])[3 * 32 + lane] = a3;
    if (lane == 0) { smx[wv] = m; sln[wv] = l; }
    __syncthreads();

    float M = -3.0e38f;
    #pragma unroll
    for (int w = 0; w < WPG; ++w) M = fmaxf(M, smx[w]);
    float e[WPG];
    float L = 0.0f;
    #pragma unroll
    for (int w = 0; w < WPG; ++w) { e[w] = __expf(smx[w] - M); L += sln[w] * e[w]; }

    const int rec = b * CPB + chunk;
    for (int h = tid; h < H_; h += 256) {
        float a = 0.0f;
        #pragma unroll
        for (int w = 0; w < WPG; ++w) a += sacc[w * H_ + h] * e[w];
        ws[WS_ACC + (size_t)rec * H_ + h] = a;
    }
    if (tid == 0) { ws[WS_M + rec] = M; ws[WS_L + rec] = L; }
}

// ---------------------------------------------------------------------------
// Kernel 2: merge the CPB partials per batch (log-sum-exp combine) and build
// combined[b][4H] = [a_mix | a_mix | det_att | output]  (b_mix aliases a_mix,
// matching the realized math of the reference).
// ---------------------------------------------------------------------------
__global__ __launch_bounds__(256)
void k2_combine(const float* __restrict__ outp, const float* __restrict__ ctx,
                const int* __restrict__ split, float* __restrict__ ws) {
    const int b   = blockIdx.x;
    const int tid = threadIdx.x;
    const int sp  = split[b];

    float M = -3.0e38f;
    #pragma unroll
    for (int i = 0; i < CPB; ++i) M = fmaxf(M, ws[WS_M + b * CPB + i]);
    float e[CPB];
    float L = 0.0f;
    #pragma unroll
    for (int i = 0; i < CPB; ++i) {
        e[i] = __expf(ws[WS_M + b * CPB + i] - M);
        L += ws[WS_L + b * CPB + i] * e[i];
    }
    const float invL = 1.0f / L;   // split_index >= 1 guarantees L > 0

    float* cb = ws + WS_COMB + (size_t)b * 4 * H_;
    for (int h = tid; h < H_; h += 256) {
        float a = 0.0f;
        #pragma unroll
        for (int i = 0; i < CPB; ++i)
            a += ws[WS_ACC + (size_t)(b * CPB + i) * H_ + h] * e[i];
        a *= invL;
        const float det = ctx[(size_t)b * S_ * H_ + (size_t)sp * H_ + h];
        const float ov  = outp[(size_t)b * H_ + h];
        cb[0 * H_ + h] = a;     // a_mix
        cb[1 * H_ + h] = a;     // b_mix (dead-code alias in reference)
        cb[2 * H_ + h] = det;   // det_att
        cb[3 * H_ + h] = ov;    // output
    }
}

// ---------------------------------------------------------------------------
// Kernel 3: out[32,512] = tanh(combined[32,2048] @ W[512,2048]^T + bias)
// via V_WMMA_F32_16X16X4_F32 (exact f32 matrix pipe). 64 waves, one 16x16
// output tile each; 512 K-steps of 4.
//   A layout: lanes 0-15 hold M=lane, VGPR0/1 = K0/K1; lanes 16-31 = K2/K3.
//   B layout: lanes 0-15 hold N=lane, VGPR0/1 = K0/K1; lanes 16-31 = K2/K3.
//   C/D:      lane, VGPR v -> M = v + (lane>=16 ? 8 : 0), N = lane%16.
// ---------------------------------------------------------------------------
__global__ __launch_bounds__(256)
void k3_gemm(const float* __restrict__ ws, const float* __restrict__ Wmat,
             const float* __restrict__ bias, float* __restrict__ out) {
    const int tid  = threadIdx.x;
    const int gw   = blockIdx.x * (256 / 32) + (tid >> 5);   // 0..63
    const int lane = tid & 31;
    const int hh   = lane >> 4;          // 0 or 1 (which half of the wave)
    const int lm   = lane & 15;
    const int m0   = (gw & 1) * 16;      // 2 M-tiles (B=32)
    const int n0   = (gw >> 1) * 16;     // 32 N-tiles (H=512)
    const int K    = 4 * H_;             // 2048

    const float* comb = ws + WS_COMB;
    const float* arow = comb + (size_t)(m0 + lm) * K;   // A[m][k]
    const float* brow = Wmat + (size_t)(n0 + lm) * K;   // B[k][n] = W[n][k]
    const int koff = hh * 2;

    v8f c = {0, 0, 0, 0, 0, 0, 0, 0};
    for (int k0 = 0; k0 < K; k0 += 4) {
        const v2f a  = *(const v2f*)(arow + k0 + koff);
        const v2f bb = *(const v2f*)(brow + k0 + koff);
        // 8 args: (neg_a, A, neg_b, B, c_mod, C, reuse_a, reuse_b)
        c = __builtin_amdgcn_wmma_f32_16x16x4_f32(
                false, a, false, bb, (short)0, c, false, false);
    }

    #pragma unroll
    for (int v = 0; v < 8; ++v) {
        const int row = m0 + v + hh * 8;
        const int col = n0 + lm;
        out[(size_t)row * H_ + col] = tanhf(c[v] + bias[col]);
    }
}

// ---------------------------------------------------------------------------
extern "C" void kernel_launch(void* const* d_in, const int* in_sizes, int n_in,
                              void* d_out, int out_size, void* d_ws, size_t ws_size,
                              hipStream_t stream) {
    const float* outp  = (const float*)d_in[0];   // [B,1,H]
    const float* ctx   = (const float*)d_in[1];   // [B,S,H]
    const int*   split = (const int*)d_in[2];     // [B]
    const float* Wmat  = (const float*)d_in[3];   // [H,4H]
    const float* bias  = (const float*)d_in[4];   // [H]

    float* out  = (float*)d_out;                  // [B,1,H] first
    float* attn = out + B_ * H_;                  // then [B,S]
    float* ws   = (float*)d_ws;                   // needs ~2.3 MB of floats

    k1_stream <<<dim3(CPB, B_), 256, 0, stream>>>(outp, ctx, split, attn, ws);
    k2_combine<<<dim3(B_),      256, 0, stream>>>(outp, ctx, split, ws);
    k3_gemm   <<<dim3(8),       256, 0, stream>>>(ws, Wmat, bias, out);
}